// SPDAttentionPool_84095459656165
// MI455X (gfx1250) — compile-verified
//
#include <hip/hip_runtime.h>
#include <math.h>

// ---------------- problem constants ----------------
#define BATCH     128
#define KDIM      256
#define SDIM      32
#define NMAT      (BATCH * KDIM)     // 32768
#define MAT_ELEMS (SDIM * SDIM)      // 1024
#define EPS_OUT   1e-4f

// Chebyshev approximation of log(x) on [CHEB_A, CHEB_B]
#define CHEB_A    0.08f
#define CHEB_B    6.0f
#define CHEB_DEG  40                 // 41 coefficients

// LDS matrix storage: 32 rows, padded stride (16B-aligned rows, conflict-free)
#define LDS_STRIDE 36
#define MAT_LDS    (SDIM * LDS_STRIDE)   // 1152 floats
#define NWAVES     4                     // waves per block in matrix kernels

// workspace layout (floats)
#define WS_CHEB   0
#define WS_QS     64
#define WS_SCORE  (WS_QS + MAT_ELEMS)            // 64 + 1024
#define WS_ALPHA  (WS_SCORE + NMAT)
#define WS_YLOG   (WS_ALPHA + NMAT)              // 128*1024 floats

typedef float v2f __attribute__((ext_vector_type(2)));
typedef float v8f __attribute__((ext_vector_type(8)));

// f32 WMMA fragment layouts (ISA 7.12.2), q = lane&15, h = lane>>4:
//   A 16x4 : row = q (+16*i), k = k0 + 2h + r   (r = vgpr 0/1)
//   B 4x16 : col = q (+16*j), k = k0 + 2h + r   (read via symmetry: B[col][k])
//   C 16x16: vgpr r -> (row = r + 8h, col = q)

// ---------------------------------------------------------------------------
// Load the 16 A-fragments (v2f each) of a staged LDS matrix into registers.
// ---------------------------------------------------------------------------
__device__ __forceinline__ void load_afrag(v2f (&aF)[2][8],
                                           const float* __restrict__ Astage,
                                           int lane) {
  const int q = lane & 15;
  const int h = lane >> 4;
#pragma unroll
  for (int i = 0; i < 2; ++i) {
    const float* Arow = Astage + (16 * i + q) * LDS_STRIDE + 2 * h;
#pragma unroll
    for (int kk = 0; kk < 8; ++kk)
      aF[i][kk] = *(const v2f*)(Arow + 4 * kk);
  }
}

// ---------------------------------------------------------------------------
// D = sScale*(A*B) [+ cScale*Cold] + diagAdd*I, A in registers (aF), B
// symmetric in LDS. HAS_COLD is compile-time (no runtime null test in the hot
// loop). Cold may alias D (per-lane read-then-write, bijective mapping).
// ---------------------------------------------------------------------------
template <bool HAS_COLD>
__device__ __forceinline__ void mm32_ra(float* D,
                                        const v2f (&aF)[2][8],
                                        const float* __restrict__ Bsym,
                                        const float* Cold,
                                        float sScale, float cScale, float diagAdd,
                                        int lane) {
  const int q = lane & 15;
  const int h = lane >> 4;
  v8f acc[4];
#pragma unroll
  for (int t = 0; t < 4; ++t) acc[t] = (v8f){};

  const float* B0 = Bsym + q * LDS_STRIDE + 2 * h;          // j = 0 fragments
  const float* B1r = Bsym + (16 + q) * LDS_STRIDE + 2 * h;  // j = 1 fragments
#pragma unroll
  for (int kk = 0; kk < 8; ++kk) {
    v2f b0 = *(const v2f*)(B0 + 4 * kk);
    v2f b1 = *(const v2f*)(B1r + 4 * kk);
    acc[0] = __builtin_amdgcn_wmma_f32_16x16x4_f32(false, aF[0][kk], false, b0,
                                                   (short)0, acc[0], false, false);
    acc[1] = __builtin_amdgcn_wmma_f32_16x16x4_f32(false, aF[0][kk], false, b1,
                                                   (short)0, acc[1], false, false);
    acc[2] = __builtin_amdgcn_wmma_f32_16x16x4_f32(false, aF[1][kk], false, b0,
                                                   (short)0, acc[2], false, false);
    acc[3] = __builtin_amdgcn_wmma_f32_16x16x4_f32(false, aF[1][kk], false, b1,
                                                   (short)0, acc[3], false, false);
  }

  // all loads precede all stores (DS ops are in-order per wave)
#pragma unroll
  for (int t = 0; t < 4; ++t) {
    const int i = t >> 1;
    const int j = t & 1;
#pragma unroll
    for (int r = 0; r < 8; ++r) {
      const int row = 16 * i + r + 8 * h;
      const int col = 16 * j + q;
      float v = sScale * acc[t][r];
      if (HAS_COLD) v += cScale * Cold[row * LDS_STRIDE + col];
      if (i == j) v += ((r + 8 * h) == q) ? diagAdd : 0.0f;  // diag tiles only
      D[row * LDS_STRIDE + col] = v;
    }
  }
}

// ---------------------------------------------------------------------------
// LDS-A variant (for squarings where A changes): D = A*B
// ---------------------------------------------------------------------------
__device__ __forceinline__ void mm32_lds(float* D,
                                         const float* __restrict__ A,
                                         const float* __restrict__ Bsym,
                                         int lane) {
  v2f aF[2][8];
  load_afrag(aF, A, lane);
  mm32_ra<false>(D, aF, Bsym, nullptr, 1.0f, 0.0f, 0.0f, lane);
}

// ---------------------------------------------------------------------------
// Matrix log of one 32x32 SPD matrix via Chebyshev-Clenshaw.
// Two LDS buffers (B1, B2); T-matrix fragments live in registers.
// Returns pointer to the LDS buffer holding log(S).
// ---------------------------------------------------------------------------
__device__ __forceinline__ float* compute_log_lds(const float* __restrict__ Sg,
                                                  float* B1, float* B2,
                                                  const float* __restrict__ cheb,
                                                  int lane) {
  const float s2 = 2.0f / (CHEB_B - CHEB_A);
  const float dd = -(CHEB_B + CHEB_A) / (CHEB_B - CHEB_A);

  // lane owns row `lane`: load 32 floats, affine map to [-1,1], stage into B1
  const float4* src = reinterpret_cast<const float4*>(Sg + lane * SDIM);
  float* trow = B1 + lane * LDS_STRIDE;
#pragma unroll
  for (int c4 = 0; c4 < 8; ++c4) {
    float4 v = src[c4];
    const int c = c4 * 4;
    trow[c + 0] = s2 * v.x + ((lane == c + 0) ? dd : 0.0f);
    trow[c + 1] = s2 * v.y + ((lane == c + 1) ? dd : 0.0f);
    trow[c + 2] = s2 * v.z + ((lane == c + 2) ? dd : 0.0f);
    trow[c + 3] = s2 * v.w + ((lane == c + 3) ? dd : 0.0f);
  }
  __builtin_amdgcn_sched_barrier(0);

  // T-matrix A-fragments -> registers (loop-invariant over all iterations)
  v2f aF[2][8];
  load_afrag(aF, B1, lane);
  __builtin_amdgcn_sched_barrier(0);

  // overwrite staging: B1 = c_D * I (b_D), B2 = 0 (b_{D+1})
  float* b1row = B1 + lane * LDS_STRIDE;
  float* b2row = B2 + lane * LDS_STRIDE;
  const float cD = cheb[CHEB_DEG];
#pragma unroll
  for (int n = 0; n < SDIM; ++n) {
    b1row[n] = (lane == n) ? cD : 0.0f;
    b2row[n] = 0.0f;
  }
  __builtin_amdgcn_sched_barrier(0);

  // Clenshaw: b_j = 2*T*b_{j+1} - b_{j+2} + c_j*I
  float* bc = B1;   // b_{j+1}
  float* bp = B2;   // b_{j+2}
#pragma unroll 1
  for (int j = CHEB_DEG - 1; j >= 1; --j) {
    mm32_ra<true>(bp, aF, bc, bp, 2.0f, -1.0f, cheb[j], lane);
    float* t = bc; bc = bp; bp = t;
    __builtin_amdgcn_sched_barrier(0);
  }
  // p = T*b1 - b2 + c0*I
  mm32_ra<true>(bp, aF, bc, bp, 1.0f, -1.0f, cheb[0], lane);
  __builtin_amdgcn_sched_barrier(0);
  return bp;
}

// ---------------------------------------------------------------------------
// K1: init — Chebyshev coefficients (device, fp64), Qs = 0.5(Q+Q^T), zero Ylog
// ---------------------------------------------------------------------------
__global__ void k_init(const float* __restrict__ Q, float* __restrict__ ws) {
  float* cheb = ws + WS_CHEB;
  float* qs   = ws + WS_QS;
  float* ylog = ws + WS_YLOG;

  if (blockIdx.x == 0) {
    const int j = threadIdx.x;
    if (j <= CHEB_DEG) {
      const int N = CHEB_DEG + 1;
      double sum = 0.0;
      for (int kk = 0; kk < N; ++kk) {
        double th = 3.14159265358979323846 * (kk + 0.5) / (double)N;
        double x  = 0.5 * ((double)CHEB_B + (double)CHEB_A) +
                    0.5 * ((double)CHEB_B - (double)CHEB_A) * cos(th);
        sum += log(x) * cos(th * (double)j);
      }
      cheb[j] = (float)(((j == 0) ? 1.0 : 2.0) * sum / (double)N);
    }
    for (int idx = threadIdx.x; idx < MAT_ELEMS; idx += blockDim.x) {
      const int i = idx >> 5, jj = idx & 31;
      qs[idx] = 0.5f * (Q[i * SDIM + jj] + Q[jj * SDIM + i]);
    }
  }
  for (int idx = blockIdx.x * blockDim.x + threadIdx.x; idx < BATCH * MAT_ELEMS;
       idx += gridDim.x * blockDim.x)
    ylog[idx] = 0.0f;
}

// ---------------------------------------------------------------------------
// K2: one wave per matrix -> log(S), Frobenius score vs Qs
// ---------------------------------------------------------------------------
__global__ void __launch_bounds__(NWAVES * 32)
k_logscore(const float* __restrict__ S, const float* __restrict__ ws_c,
           float* __restrict__ ws) {
  __shared__ __align__(16) float lds[NWAVES * 2 * MAT_LDS];
  const int wid  = threadIdx.x >> 5;
  const int lane = threadIdx.x & 31;
  const int m    = blockIdx.x * NWAVES + wid;     // flat matrix index

  float* B1 = lds + wid * 2 * MAT_LDS;
  float* B2 = B1 + MAT_LDS;

  const float* cheb = ws_c + WS_CHEB;
  const float* qs   = ws_c + WS_QS;
  float* scores     = ws + WS_SCORE;

  const float* Lp = compute_log_lds(S + (size_t)m * MAT_ELEMS, B1, B2, cheb, lane);

  // score = -||L - Qs||_F^2 / 32 ; lane sums its row, then wave-reduce
  const float* lrow = Lp + lane * LDS_STRIDE;
  const float* qrow = qs + lane * SDIM;
  float acc = 0.0f;
#pragma unroll
  for (int n = 0; n < SDIM; ++n) {
    const float d = lrow[n] - qrow[n];
    acc += d * d;
  }
#pragma unroll
  for (int off = 16; off >= 1; off >>= 1) acc += __shfl_xor(acc, off, 32);
  if (lane == 0) scores[m] = -acc * (1.0f / (float)SDIM);
}

// ---------------------------------------------------------------------------
// K3: softmax over K=256 per batch. grid 128 x 256
// ---------------------------------------------------------------------------
__global__ void k_softmax(float* __restrict__ ws) {
  __shared__ float red[KDIM];
  const float* scores = ws + WS_SCORE;
  float* alphav       = ws + WS_ALPHA;
  const int b = blockIdx.x, t = threadIdx.x;

  const float s = scores[b * KDIM + t];
  red[t] = s;
  __syncthreads();
  for (int off = KDIM / 2; off >= 1; off >>= 1) {
    if (t < off) red[t] = fmaxf(red[t], red[t + off]);
    __syncthreads();
  }
  const float mx = red[0];
  __syncthreads();
  const float e = expf(s - mx);
  red[t] = e;
  __syncthreads();
  for (int off = KDIM / 2; off >= 1; off >>= 1) {
    if (t < off) red[t] = red[t] + red[t + off];
    __syncthreads();
  }
  alphav[b * KDIM + t] = e / red[0];
}

// ---------------------------------------------------------------------------
// K4: recompute logs, accumulate alpha-weighted sum into Ylog[b].
// One wave handles 16 consecutive k of one batch.
// ---------------------------------------------------------------------------
__global__ void __launch_bounds__(NWAVES * 32)
k_accum(const float* __restrict__ S, const float* __restrict__ ws_c,
        float* __restrict__ ws) {
  __shared__ __align__(16) float lds[NWAVES * 2 * MAT_LDS];
  const int wid  = threadIdx.x >> 5;
  const int lane = threadIdx.x & 31;
  const int w    = blockIdx.x * NWAVES + wid;     // 0..2047
  const int b    = w >> 4;

  float* B1 = lds + wid * 2 * MAT_LDS;
  float* B2 = B1 + MAT_LDS;

  const float* cheb   = ws_c + WS_CHEB;
  const float* alphav = ws_c + WS_ALPHA;
  float* ylog         = ws + WS_YLOG;

  float yacc[SDIM];
#pragma unroll
  for (int n = 0; n < SDIM; ++n) yacc[n] = 0.0f;

#pragma unroll 1
  for (int kk = 0; kk < 16; ++kk) {
    const int m = w * 16 + kk;                    // = b*256 + k
    const float* Lp = compute_log_lds(S + (size_t)m * MAT_ELEMS, B1, B2, cheb, lane);
    const float aw = alphav[m];
    const float* lrow = Lp + lane * LDS_STRIDE;
#pragma unroll
    for (int n = 0; n < SDIM; ++n) yacc[n] += aw * lrow[n];
    __builtin_amdgcn_sched_barrier(0);
  }

  float* dst = ylog + (size_t)b * MAT_ELEMS + lane * SDIM;
#pragma unroll
  for (int n = 0; n < SDIM; ++n) atomicAdd(dst + n, yacc[n]);
}

// ---------------------------------------------------------------------------
// K5: Y = exp(Ylog) + EPS*I via scaling-and-squaring (X/16, Taylor deg 8, ^16).
// One wave per batch.
// ---------------------------------------------------------------------------
__global__ void __launch_bounds__(NWAVES * 32)
k_expout(const float* __restrict__ ws_c, float* __restrict__ out) {
  __shared__ __align__(16) float lds[NWAVES * 2 * MAT_LDS];
  const int wid  = threadIdx.x >> 5;
  const int lane = threadIdx.x & 31;
  const int b    = blockIdx.x * NWAVES + wid;     // 0..127

  float* B1 = lds + wid * 2 * MAT_LDS;
  float* B2 = B1 + MAT_LDS;

  const float* ylog = ws_c + WS_YLOG;
  const float* yrow = ylog + (size_t)b * MAT_ELEMS + lane * SDIM;

  // stage X = Ylog / 2^4 into B1, pull A-fragments to registers
  float* trow = B1 + lane * LDS_STRIDE;
#pragma unroll
  for (int n = 0; n < SDIM; ++n) trow[n] = yrow[n] * (1.0f / 16.0f);
  __builtin_amdgcn_sched_barrier(0);

  v2f aF[2][8];
  load_afrag(aF, B1, lane);
  __builtin_amdgcn_sched_barrier(0);

  float* b1row = B1 + lane * LDS_STRIDE;
#pragma unroll
  for (int n = 0; n < SDIM; ++n) b1row[n] = (lane == n) ? 1.0f : 0.0f;  // P = I
  __builtin_amdgcn_sched_barrier(0);

  float* P  = B1;
  float* Wk = B2;
#pragma unroll 1
  for (int j = 8; j >= 1; --j) {                  // P = I + (X*P)/j
    mm32_ra<false>(Wk, aF, P, nullptr, 1.0f / (float)j, 0.0f, 1.0f, lane);
    float* t = P; P = Wk; Wk = t;
    __builtin_amdgcn_sched_barrier(0);
  }
#pragma unroll 1
  for (int sq = 0; sq < 4; ++sq) {                // P = P*P, four times
    mm32_lds(Wk, P, P, lane);
    float* t = P; P = Wk; Wk = t;
    __builtin_amdgcn_sched_barrier(0);
  }

  float* orow = out + (size_t)b * MAT_ELEMS + lane * SDIM;
  const float* prow = P + lane * LDS_STRIDE;
#pragma unroll
  for (int n = 0; n < SDIM; ++n)
    orow[n] = prow[n] + ((lane == n) ? EPS_OUT : 0.0f);
}

// ---------------------------------------------------------------------------
extern "C" void kernel_launch(void* const* d_in, const int* in_sizes, int n_in,
                              void* d_out, int out_size, void* d_ws, size_t ws_size,
                              hipStream_t stream) {
  (void)in_sizes; (void)n_in; (void)out_size; (void)ws_size;
  const float* S = (const float*)d_in[0];   // [128,256,32,32]
  const float* Q = (const float*)d_in[1];   // [32,32]
  float* out = (float*)d_out;               // [128,32,32]
  float* ws  = (float*)d_ws;

  k_init    <<<128,  256, 0, stream>>>(Q, ws);
  k_logscore<<<NMAT / NWAVES, NWAVES * 32, 0, stream>>>(S, ws, ws);
  k_softmax <<<BATCH, KDIM, 0, stream>>>(ws);
  k_accum   <<<(NMAT / 16) / NWAVES, NWAVES * 32, 0, stream>>>(S, ws, ws);
  k_expout  <<<BATCH / NWAVES, NWAVES * 32, 0, stream>>>(ws, out);
}